// VectorQuantizer_44169443672274
// MI455X (gfx1250) — compile-verified
//
#include <hip/hip_runtime.h>

typedef float v2f __attribute__((ext_vector_type(2)));
typedef float v8f __attribute__((ext_vector_type(8)));

#define NBATCH 16
#define CH     8
#define HH     64
#define WW     128
#define HW     (HH * WW)          // 8192
#define CHW    (CH * HW)          // 65536
#define NTOT   (NBATCH * HW)      // 131072 points
#define KCODES 2048
#define ZQ_N   (NBATCH * CH * HW) // 1048576 floats of z_q_st
#define LOSS_OFF ZQ_N             // scalar loss
#define IDX_OFF  (ZQ_N + 1)       // 131072 indices (stored as float)
#define ZBASE  (KCODES + 16)      // padded zero region (bank-shifted by 16 floats)

// 256 threads = 8 wave32s; each wave owns one 16-point tile; grid = 131072/16/8 = 1024
__global__ __launch_bounds__(256) void vq_wmma_kernel(
    const float* __restrict__ z, const float* __restrict__ emb,
    float* __restrict__ out, float* __restrict__ partial)
{
    // [0,2048): -0.5*||emb_k||^2 ; [2064, 4112): zeros for B rows K=10/11
    __shared__ float s_bscale[ZBASE + KCODES];
    __shared__ int   s_bidx[8][16];     // per-wave winning code per tile row
    __shared__ float s_loss[8];         // per-wave loss partial

    const int tid  = threadIdx.x;
    const int wave = tid >> 5;
    const int lane = tid & 31;
    const int l15  = lane & 15;

    // ---- 1) -0.5 * codebook norms into LDS (coalesced b128 loads) + zero pad ----
    #pragma unroll
    for (int i = 0; i < 8; ++i) {
        int code = i * 256 + tid;
        const float4* ep = (const float4*)(emb + code * 8);
        float4 e0 = ep[0], e1 = ep[1];
        s_bscale[code] = -0.5f * (e0.x*e0.x + e0.y*e0.y + e0.z*e0.z + e0.w*e0.w +
                                  e1.x*e1.x + e1.y*e1.y + e1.z*e1.z + e1.w*e1.w);
    }
    for (int i = KCODES + tid; i < ZBASE + KCODES; i += 256)
        s_bscale[i] = 0.0f;
    __syncthreads();

    // ---- 2) A fragments: 16x8 tile of z_flat as two 16x4 f32 A-matrices,
    //         plus constant A2 = e_9th-column (ones) for the norm-folding WMMA ----
    // A layout (ISA 7.12.2): lanes 0-15 = rows M, VGPR0={K0|K2}, VGPR1={K1|K3}
    const int tileRow = (blockIdx.x * 8 + wave) * 16;
    const int row = tileRow + l15;
    const int bb  = row >> 13;          // row / HW
    const int rr  = row & (HW - 1);     // h*W + w
    const long base = (long)bb * CHW + rr;
    const int c0 = (lane < 16) ? 0 : 2;
    v2f a0, a1, a2;
    a0.x = z[base + (long)(c0 + 0) * HW];
    a0.y = z[base + (long)(c0 + 1) * HW];
    a1.x = z[base + (long)(c0 + 4) * HW];
    a1.y = z[base + (long)(c0 + 5) * HW];
    a2.x = (lane < 16) ? 1.0f : 0.0f;   // A[m][8] = 1, A[m][9..11] = 0
    a2.y = 0.0f;

    // per-lane B2 source: lanes 0-15 read -0.5*||e||^2 (B row K=8), lanes 16-31 read zeros
    const float* bsel = s_bscale + ((lane < 16) ? 0 : ZBASE);

    float best[8];
    int   bidx[8];
    #pragma unroll
    for (int j = 0; j < 8; ++j) { best[j] = -3.4e38f; bidx[j] = 0; }

    // ---- 3) sweep the codebook, 32 codes per trip (2 chunks interleaved) ----
    // acc = z.e - 0.5*||e||^2  ==> argmax(acc) == argmin(||z-e||^2)
    for (int cb = 0; cb < KCODES; cb += 32) {
        #pragma unroll
        for (int u = 0; u < 2; ++u) {
            const int n = cb + u * 16 + l15;
            // B mirrors A: lane<16 holds emb[n][0..1], lane>=16 holds emb[n][2..3]
            const v2f* bp = (const v2f*)(emb + n * 8 + c0);
            v2f b0 = bp[0];                      // channels c0, c0+1
            v2f b1 = bp[2];                      // channels c0+4, c0+5
            v2f b2; b2.x = bsel[n]; b2.y = 0.0f; // B[8][n] = -0.5*||e_n||^2

            v8f acc = {};
            acc = __builtin_amdgcn_wmma_f32_16x16x4_f32(false, a0, false, b0,
                                                        (short)0, acc, false, false);
            acc = __builtin_amdgcn_wmma_f32_16x16x4_f32(false, a1, false, b1,
                                                        (short)0, acc, false, false);
            acc = __builtin_amdgcn_wmma_f32_16x16x4_f32(false, a2, false, b2,
                                                        (short)0, acc, false, false);
            #pragma unroll
            for (int j = 0; j < 8; ++j) {
                if (acc[j] > best[j]) { best[j] = acc[j]; bidx[j] = n; }
            }
        }
    }

    // ---- 4) argmax across the 16-lane column groups (xor stays inside group) ----
    #pragma unroll
    for (int j = 0; j < 8; ++j) {
        #pragma unroll
        for (int off = 8; off >= 1; off >>= 1) {
            float ob = __shfl_xor(best[j], off);
            int   oi = __shfl_xor(bidx[j], off);
            if (ob > best[j] || (ob == best[j] && oi < bidx[j])) {
                best[j] = ob; bidx[j] = oi;   // tie -> lowest index (matches argmin)
            }
        }
    }
    if (l15 == 0) {
        const int mbase = (lane < 16) ? 0 : 8;  // D layout: lanes16-31 are rows M=j+8
        #pragma unroll
        for (int j = 0; j < 8; ++j) s_bidx[wave][mbase + j] = bidx[j];
    }
    __syncthreads();

    // ---- 5) gather codebook rows, write z_q_st + idx, accumulate loss ----
    float acc2 = 0.0f;
    if (lane < 16) {
        const int idxv = s_bidx[wave][lane];
        const int row2 = tileRow + lane;
        out[IDX_OFF + row2] = (float)idxv;
        const int b2i = row2 >> 13;
        const int r2  = row2 & (HW - 1);
        const long ob2 = (long)b2i * CHW + r2;
        const float4* ep = (const float4*)(emb + idxv * 8);
        float4 e0 = ep[0], e1 = ep[1];
        float ev[8] = { e0.x, e0.y, e0.z, e0.w, e1.x, e1.y, e1.z, e1.w };
        #pragma unroll
        for (int c = 0; c < 8; ++c) {
            long a = ob2 + (long)c * HW;
            float zv = z[a];
            float d = ev[c] - zv;
            out[a] = zv + d;          // z + sg(z_q - z): bit-exact forward
            acc2 += d * d;
        }
    }
    #pragma unroll
    for (int off = 16; off >= 1; off >>= 1)
        acc2 += __shfl_xor(acc2, off);
    if (lane == 0) s_loss[wave] = acc2;
    __syncthreads();
    if (tid == 0) {
        float t = 0.0f;
        #pragma unroll
        for (int w = 0; w < 8; ++w) t += s_loss[w];
        partial[blockIdx.x] = t;      // plain store: deterministic, no atomics
    }
}

__global__ __launch_bounds__(256) void vq_loss_reduce(
    const float* __restrict__ partial, float* __restrict__ out)
{
    __shared__ float s[8];
    const int tid = threadIdx.x;
    float a = 0.0f;
    #pragma unroll
    for (int i = 0; i < 4; ++i) a += partial[i * 256 + tid];
    #pragma unroll
    for (int off = 16; off >= 1; off >>= 1) a += __shfl_xor(a, off);
    if ((tid & 31) == 0) s[tid >> 5] = a;
    __syncthreads();
    if (tid == 0) {
        float t = 0.0f;
        #pragma unroll
        for (int i = 0; i < 8; ++i) t += s[i];
        // loss = mean((zq-z)^2) + 0.25*mean((zq-z)^2)
        out[LOSS_OFF] = 1.25f * t / (float)ZQ_N;
    }
}

extern "C" void kernel_launch(void* const* d_in, const int* in_sizes, int n_in,
                              void* d_out, int out_size, void* d_ws, size_t ws_size,
                              hipStream_t stream) {
    const float* z   = (const float*)d_in[0];   // [16,8,64,128] f32
    const float* emb = (const float*)d_in[1];   // [2048,8] f32
    float* out     = (float*)d_out;             // [zq 1048576][loss 1][idx 131072]
    float* partial = (float*)d_ws;              // 1024 floats scratch
    vq_wmma_kernel<<<dim3(NTOT / 16 / 8), dim3(256), 0, stream>>>(z, emb, out, partial);
    vq_loss_reduce<<<dim3(1), dim3(256), 0, stream>>>(partial, out);
}